// CA_Module_16836271800509
// MI455X (gfx1250) — compile-verified
//
#include <hip/hip_runtime.h>
#include <hip/hip_bf16.h>

// CA_Module channel self-attention for MI455X (gfx1250), fp32 end-to-end.
// b=16, c=64, N=65536. Bandwidth-bound (~0.8 GB HBM, ~35us floor @23.3TB/s).
// Matrix math: V_WMMA_F32_16X16X4_F32 (full fp32 precision).
// Data movement: Tensor Data Mover (tensor_load_to_lds) double-buffered DMA
// with hardware LDS row padding; non-temporal stores for the write-only out.

typedef __attribute__((ext_vector_type(2))) float v2f;
typedef __attribute__((ext_vector_type(4))) float v4f;
typedef __attribute__((ext_vector_type(8))) float v8f;
typedef __attribute__((ext_vector_type(4))) unsigned int v4u;
typedef __attribute__((ext_vector_type(4))) int v4i;
typedef __attribute__((ext_vector_type(8))) int v8i;

#define BATCHES 16
#define CH      64
#define NPIX    65536          // 256*256
#define S1      68             // gram LDS row stride (64 + 4 pad dwords)
#define S3      76             // apply LDS row stride (64 + 12 pad dwords)

#if defined(__has_builtin)
#if __has_builtin(__builtin_amdgcn_tensor_load_to_lds) && \
    __has_builtin(__builtin_amdgcn_s_wait_tensorcnt)
#define CA_HAS_TDM 1
#endif
#endif

// The shared aperture maps the LDS byte offset into the low 32 bits of the
// generic (flat) address (ISA 10.2), so truncation
// yields the D# lds_addr for the TDM descriptor.
__device__ __forceinline__ unsigned ca_lds_off(const void* p) {
    return (unsigned)(unsigned long long)(size_t)p;
}

// The TDM DMA writes LDS behind the compiler's back (the descriptor carries
// the address as an integer). Escape the pointer (input-only!) with a memory
// clobber: the compiler must assume the buffer was written (round 2: loads
// folded to undef), but the pointer VALUE is untouched so its AS(3) provenance
// survives and reads stay ds_load (round 3's "+v" rewrite degraded them to
// flat_load_b64 through the generic aperture).
__device__ __forceinline__ void ca_tdm_fence(const void* p) {
    asm volatile("" : : "v"(p) : "memory");
}

#ifdef CA_HAS_TDM
// Issue a TDM DMA of a 64x64 fp32 tile (row stride NPIX floats) from global
// into LDS at byte offset lds_off, with hardware padding of `padCode`+1
// dwords after every 64 dwords (reproduces the padded LDS strides S1/S3).
// D# layout per CDNA5 ISA ch.10.7/§8 (group0: count/lds/global/type,
// group1: data_size|pad|dims|strides, groups 2-3 zero for 2D tiles).
__device__ __forceinline__ void ca_tdm_tile64(unsigned lds_off,
                                              const float* gsrc,
                                              unsigned padCode) {
    const unsigned long long ga = (unsigned long long)gsrc;
    v4u g0;
    g0[0] = 1u;                                   // count=1, user descriptor
    g0[1] = lds_off;                              // lds_addr (bytes)
    g0[2] = (unsigned)(ga & 0xFFFFFFFFu);         // global_addr[31:0]
    g0[3] = (unsigned)((ga >> 32) & 0x01FFFFFFu)  // global_addr[56:32]
          | (2u << 30);                           // type=2 ("image")
    v8i g1;
    g1[0] = (int)((2u << 16)                      // data_size = 4B
                | (1u << 20)                      // pad_enable
                | (5u << 22)                      // pad_interval: 64 dwords
                | (padCode << 25));               // pad_amount code
    g1[1] = (int)(64u << 16);                     // tensor_dim0 = 64 (lo16)
    g1[2] = (int)(64u << 16);                     // dim0 hi=0 | tensor_dim1 = 64
    g1[3] = (int)(64u << 16);                     // dim1 hi=0 | tile_dim0 = 64
    g1[4] = 64;                                   // tile_dim1=64, tile_dim2=0
    g1[5] = (int)NPIX;                            // tensor_dim0_stride lo32
    g1[6] = 0;                                    // stride0 hi | stride1 lo
    g1[7] = 0;                                    // stride1 hi
    v4i gz = {0, 0, 0, 0};
#if defined(__clang_major__) && __clang_major__ >= 23
    v8i gz8 = {0, 0, 0, 0, 0, 0, 0, 0};
    __builtin_amdgcn_tensor_load_to_lds(g0, g1, gz, gz, gz8, 0);
#else
    __builtin_amdgcn_tensor_load_to_lds(g0, g1, gz, gz, 0);
#endif
}
#endif // CA_HAS_TDM

// ---------------------------------------------------------------- zero ws
__global__ __launch_bounds__(256) void ca_zero_kernel(float* __restrict__ p, int n) {
    int i = blockIdx.x * 256 + threadIdx.x;
    if (i < n) p[i] = 0.0f;
}

// ------------------------------------------------ gram tile compute (WMMA)
// A layout (16x4 f32): lane holds M=l15; VGPR0=K+0/+2 (by half), VGPR1=K+1/+3.
// B (Gram: B(k,d)=q[d,k]) mirrors A's lane pattern.
__device__ __forceinline__ void ca_gram_compute(const float* L,
                                                int ct0, int dt0, int ct1, int dt1,
                                                int l15, int hi,
                                                v8f& acc0, v8f& acc1) {
    #pragma unroll
    for (int kk = 0; kk < 64; kk += 4) {
        const int cofs = kk + 2 * hi;
        v2f a0 = *(const v2f*)(&L[(ct0 * 16 + l15) * S1 + cofs]);
        v2f b0 = *(const v2f*)(&L[(dt0 * 16 + l15) * S1 + cofs]);
        acc0 = __builtin_amdgcn_wmma_f32_16x16x4_f32(false, a0, false, b0,
                                                     (short)0, acc0, false, false);
        v2f a1 = *(const v2f*)(&L[(ct1 * 16 + l15) * S1 + cofs]);
        v2f b1 = *(const v2f*)(&L[(dt1 * 16 + l15) * S1 + cofs]);
        acc1 = __builtin_amdgcn_wmma_f32_16x16x4_f32(false, a1, false, b1,
                                                     (short)0, acc1, false, false);
    }
}

// ------------------------------------------------ K1: att += q_chunk * q^T
// grid = (16 batches, 32 chunks of 2048 cols), block = 256 (8 waves).
// TDM double-buffered 64x64 LDS tiles; wave w owns 16x16 att tiles w, w+8.
__global__ __launch_bounds__(256) void ca_gram_kernel(const float* __restrict__ x,
                                                      float* __restrict__ att_acc) {
    __shared__ float lds[2][CH * S1];

    const int b    = blockIdx.x;
    const int tid  = threadIdx.x;
    const int wave = tid >> 5;
    const int lane = tid & 31;
    const int l15  = lane & 15;
    const int hi   = lane >> 4;

    const float* xb0 = x + (size_t)b * CH * NPIX + blockIdx.y * 2048;

    const int ct0 = (wave    ) >> 2, dt0 = (wave    ) & 3;
    const int ct1 = (wave + 8) >> 2, dt1 = (wave + 8) & 3;

    v8f acc0 = {};
    v8f acc1 = {};
    const int nTiles = 32;                    // 2048 / 64

#ifdef CA_HAS_TDM
    const unsigned ldsOff[2] = { ca_lds_off(&lds[0][0]), ca_lds_off(&lds[1][0]) };
    if (tid < 32) ca_tdm_tile64(ldsOff[0], xb0, 3u);   // tile 0, pad 4 dwords
    for (int t = 0; t < nTiles; ++t) {
        if (tid < 32) __builtin_amdgcn_s_wait_tensorcnt(0);
        __syncthreads();                      // tile t visible; buf[(t+1)&1] free
        ca_tdm_fence(&lds[t & 1][0]);         // DMA wrote this buffer
        if (tid < 32 && t + 1 < nTiles)
            ca_tdm_tile64(ldsOff[(t + 1) & 1], xb0 + (t + 1) * 64, 3u);
        ca_gram_compute(&lds[t & 1][0], ct0, dt0, ct1, dt1, l15, hi, acc0, acc1);
    }
#else
    for (int t = 0; t < nTiles; ++t) {
        __syncthreads();
        #pragma unroll
        for (int it = 0; it < 4; ++it) {      // 64x64 tile, coalesced float4
            int idx = it * 256 + tid;
            int row = idx >> 4;
            int col = (idx & 15) * 4;
            v4f v = *(const v4f*)(xb0 + (size_t)row * NPIX + t * 64 + col);
            *(v4f*)(&lds[0][row * S1 + col]) = v;
        }
        __syncthreads();
        ca_gram_compute(&lds[0][0], ct0, dt0, ct1, dt1, l15, hi, acc0, acc1);
    }
#endif

    // C/D layout: VGPR r -> row r (lanes 0-15) / row r+8 (lanes 16-31).
    float* attb = att_acc + b * CH * CH;
    #pragma unroll
    for (int r = 0; r < 8; ++r) {
        atomicAdd(&attb[(ct0 * 16 + r + 8 * hi) * CH + dt0 * 16 + l15], acc0[r]);
        atomicAdd(&attb[(ct1 * 16 + r + 8 * hi) * CH + dt1 * 16 + l15], acc1[r]);
    }
}

// ------------------------------------------------ K2: reversed softmax
// softmax(rowmax - att) = exp(min(att) - att) / sum  (stable closed form).
__global__ __launch_bounds__(64) void ca_softmax_kernel(const float* __restrict__ att_acc,
                                                        float* __restrict__ att_sm) {
    const int b = blockIdx.x;
    const int r = threadIdx.x;
    const float* row = att_acc + (b * CH + r) * CH;
    float*       dst = att_sm  + (b * CH + r) * CH;

    float mn = row[0];
    #pragma unroll 8
    for (int i = 1; i < CH; ++i) mn = fminf(mn, row[i]);

    float sum = 0.0f;
    #pragma unroll 8
    for (int i = 0; i < CH; ++i) sum += __expf(mn - row[i]);

    const float inv = 1.0f / sum;
    #pragma unroll 8
    for (int i = 0; i < CH; ++i) dst[i] = __expf(mn - row[i]) * inv;
}

// ------------------------------------------------ apply tile compute (WMMA)
__device__ __forceinline__ void ca_apply_tile(const float* L, const v2f* aReg,
                                              int ct, int nhalf, int l15, int hi,
                                              float g, float* ob, int n0) {
    #pragma unroll
    for (int jj = 0; jj < 2; ++jj) {
        const int nloc = (nhalf * 2 + jj) * 16;
        v8f acc = {};
        #pragma unroll
        for (int k = 0; k < 16; ++k) {
            const int r0 = 4 * k + 2 * hi;       // B(k,n): K row per half-wave
            v2f bv;
            bv.x = L[(r0    ) * S3 + nloc + l15];
            bv.y = L[(r0 + 1) * S3 + nloc + l15];
            acc = __builtin_amdgcn_wmma_f32_16x16x4_f32(false, aReg[k], false, bv,
                                                        (short)0, acc, false, false);
        }
        #pragma unroll
        for (int r = 0; r < 8; ++r) {
            const int row = ct * 16 + r + 8 * hi;
            const int col = nloc + l15;
            const float xv = L[row * S3 + col];  // residual from LDS
            __builtin_nontemporal_store(g * acc[r] + xv,
                                        &ob[(size_t)row * NPIX + n0 + col]);
        }
    }
}

// ------------------------------------------------ K3: out = gamma*(att_sm@q)+x
// grid = (16 batches, 64 ranges of 1024 cols), block = 256 (8 waves).
// Wave: fixed c-tile ct=w&3, n-half=w>>2; A preloaded in 16 v2f registers.
__global__ __launch_bounds__(256) void ca_apply_kernel(const float* __restrict__ x,
                                                       const float* __restrict__ att_sm,
                                                       const float* __restrict__ gamma,
                                                       float* __restrict__ out) {
    __shared__ float lds[2][CH * S3];

    const int b    = blockIdx.x;
    const int tid  = threadIdx.x;
    const int wave = tid >> 5;
    const int lane = tid & 31;
    const int l15  = lane & 15;
    const int hi   = lane >> 4;

    const int ct    = wave & 3;
    const int nhalf = wave >> 2;
    const float g   = gamma[0];

    const int    nbase = blockIdx.y * 1024;
    const float* xb0   = x   + (size_t)b * CH * NPIX + nbase;
    float*       ob    = out + (size_t)b * CH * NPIX;

    v2f aReg[16];
    {
        const float* arow = att_sm + (b * CH + ct * 16 + l15) * CH;
        #pragma unroll
        for (int k = 0; k < 16; ++k)
            aReg[k] = *(const v2f*)(arow + 4 * k + 2 * hi);
    }

    const int nTiles = 16;                    // 1024 / 64

#ifdef CA_HAS_TDM
    const unsigned ldsOff[2] = { ca_lds_off(&lds[0][0]), ca_lds_off(&lds[1][0]) };
    if (tid < 32) ca_tdm_tile64(ldsOff[0], xb0, 11u);  // pad 12 dw -> stride 76
    for (int t = 0; t < nTiles; ++t) {
        if (tid < 32) __builtin_amdgcn_s_wait_tensorcnt(0);
        __syncthreads();
        ca_tdm_fence(&lds[t & 1][0]);
        if (tid < 32 && t + 1 < nTiles)
            ca_tdm_tile64(ldsOff[(t + 1) & 1], xb0 + (t + 1) * 64, 11u);
        ca_apply_tile(&lds[t & 1][0], aReg, ct, nhalf, l15, hi, g, ob,
                      nbase + t * 64);
    }
#else
    for (int t = 0; t < nTiles; ++t) {
        __syncthreads();
        #pragma unroll
        for (int it = 0; it < 4; ++it) {
            int idx = it * 256 + tid;
            int row = idx >> 4;
            int col = (idx & 15) * 4;
            v4f v = *(const v4f*)(xb0 + (size_t)row * NPIX + t * 64 + col);
            *(v4f*)(&lds[0][row * S3 + col]) = v;
        }
        __syncthreads();
        ca_apply_tile(&lds[0][0], aReg, ct, nhalf, l15, hi, g, ob,
                      nbase + t * 64);
    }
#endif
}

// ---------------------------------------------------------------- launcher
extern "C" void kernel_launch(void* const* d_in, const int* in_sizes, int n_in,
                              void* d_out, int out_size, void* d_ws, size_t ws_size,
                              hipStream_t stream) {
    const float* x     = (const float*)d_in[0];
    const float* gamma = (const float*)d_in[1];
    float*       out   = (float*)d_out;

    // Workspace: [att_acc : 16*64*64 f32][att_sm : 16*64*64 f32] = 512 KB.
    float* att_acc = (float*)d_ws;
    float* att_sm  = att_acc + BATCHES * CH * CH;

    const int attElems = BATCHES * CH * CH;
    ca_zero_kernel<<<(attElems + 255) / 256, 256, 0, stream>>>(att_acc, attElems);

    dim3 g1(BATCHES, NPIX / 2048);   // 16 x 32 WGs
    ca_gram_kernel<<<g1, 256, 0, stream>>>(x, att_acc);

    ca_softmax_kernel<<<BATCHES, 64, 0, stream>>>(att_acc, att_sm);

    dim3 g3(BATCHES, NPIX / 1024);   // 16 x 64 WGs
    ca_apply_kernel<<<g3, 256, 0, stream>>>(x, att_sm, gamma, out);
}